// ScaledDotProductAttention_33543694581833
// MI455X (gfx1250) — compile-verified
//
#include <hip/hip_runtime.h>
#include <stdint.h>

// CDNA5 (gfx1250) wave32 WMMA types
typedef __attribute__((ext_vector_type(2))) float v2f;   // f32 A/B 16x4 / 4x16 tile: 2 VGPRs
typedef __attribute__((ext_vector_type(8))) float v8f;   // f32 C/D 16x16 tile: 8 VGPRs

namespace {
constexpr int kB  = 16;
constexpr int kLq = 256;
constexpr int kLk = 256;
constexpr int kDk = 64;
constexpr int kQT = 16;  // q rows per workgroup (WMMA M)
constexpr float kInvTemper = 0.04419417382415922f;  // 1/sqrt(D_MODEL=512)

__device__ __forceinline__ float hsum16(float v) {
  // butterfly reduction within a 16-lane half of a wave32
  v += __shfl_xor(v, 8, 16);
  v += __shfl_xor(v, 4, 16);
  v += __shfl_xor(v, 2, 16);
  v += __shfl_xor(v, 1, 16);
  return v;
}
__device__ __forceinline__ float hmax16(float v) {
  v = fmaxf(v, __shfl_xor(v, 8, 16));
  v = fmaxf(v, __shfl_xor(v, 4, 16));
  v = fmaxf(v, __shfl_xor(v, 2, 16));
  v = fmaxf(v, __shfl_xor(v, 1, 16));
  return v;
}
}  // namespace

// One workgroup = one (batch b, q-tile of 16 rows). 256 threads = 8 wave32.
__global__ __launch_bounds__(256)
void attn_pos_kernel(const float* __restrict__ q_g,      // [B][Lq][Dk]
                     const float* __restrict__ k_g,      // [B][Lk][Dk]
                     const float* __restrict__ v_g,      // [B][Lk][Dk]
                     const uint8_t* __restrict__ mask_g, // [B][Lq][Lk] (bool)
                     const float* __restrict__ pos_g,    // [B][Lq][Lk][2*Dk]
                     float* __restrict__ out_g,          // [B][Lq][Dk]
                     float* __restrict__ attn_g) {       // [B][Lq][Lk]
  // Row pads: 257 -> bank+1 per row (kills conflicts on column reads of A),
  // 68 floats = 272B -> rows stay 16B aligned for float4, bank+4 per row.
  __shared__ float sQ[kQT][68];
  __shared__ float sS[kQT][257];
  __shared__ float sO[kQT][68];

  const int qt = blockIdx.x;
  const int b  = blockIdx.y;
  const int q0 = qt * kQT;
  const int tid  = threadIdx.x;
  const int lane = tid & 31;
  const int wave = tid >> 5;
  const int half = lane >> 4;  // which 16-lane half of the wave
  const int sub  = lane & 15;

  // ---- load Q tile into LDS (coalesced float4), zero output accumulator ----
  {
    const int row = tid >> 4;
    const int col = (tid & 15) * 4;
    const float4 qv =
        *(const float4*)&q_g[((size_t)b * kLq + q0 + row) * kDk + col];
    *(float4*)&sQ[row][col] = qv;
  }
  for (int idx = tid; idx < kQT * 68; idx += 256) (&sO[0][0])[idx] = 0.f;
  __syncthreads();

  // ================= Phase A: scores =================
  // s[i][k] = (q_i . k_k + q_i . kpos[b,q0+i,k,:]) / temper
  //         = (q_i . (k_k + kpos_row)) / temper
  // Half-wave cooperates on one dot: 16 lanes x float4 = 256B coalesced.
  // Streams the k_pos half of pos_emb exactly once (HBM-bound part #1).
  for (int s = 0; s < 256; ++s) {
    const int p  = s * 16 + wave * 2 + half;  // bijection over 16x256 pairs
    const int i  = p >> 8;
    const int kk = p & 255;
    const float4 qv = *(const float4*)&sQ[i][sub * 4];
    const float4 kv =
        *(const float4*)&k_g[((size_t)b * kLk + kk) * kDk + sub * 4];
    const float* prow =
        pos_g + (((size_t)b * kLq + q0 + i) * kLk + kk) * (2 * kDk);
    const float4 pv = *(const float4*)&prow[sub * 4];
    __builtin_prefetch(prow + 16 * (2 * kDk) + sub * 4, 0, 0);  // next k for this wave
    float part = qv.x * (kv.x + pv.x) + qv.y * (kv.y + pv.y) +
                 qv.z * (kv.z + pv.z) + qv.w * (kv.w + pv.w);
    part = hsum16(part);
    if (sub == 0) sS[i][kk] = part * kInvTemper;
  }
  __syncthreads();

  // ================= Phase B: mask + softmax + attn output =================
  // Each half-wave owns one of the 16 rows; each lane owns 16 columns.
  {
    const int row = wave * 2 + half;
    const size_t rbase = ((size_t)b * kLq + q0 + row) * kLk;
    const uint8_t* mrow = mask_g + rbase;
    float sv[16];
    float rowmax = -INFINITY;
#pragma unroll
    for (int j = 0; j < 16; ++j) {
      const int c = sub + j * 16;
      float s = sS[row][c];
      if (mrow[c]) s = -INFINITY;  // mask==True -> -inf
      sv[j] = s;
      rowmax = fmaxf(rowmax, s);
    }
    rowmax = hmax16(rowmax);
    if (rowmax == -INFINITY) {
      // fully-masked row: softmax is NaN -> reference maps to zeros
#pragma unroll
      for (int j = 0; j < 16; ++j) sv[j] = 0.f;
    } else {
      float denom = 0.f;
#pragma unroll
      for (int j = 0; j < 16; ++j) {
        sv[j] = __expf(sv[j] - rowmax);  // exp(-inf - m) = 0 for masked
        denom += sv[j];
      }
      denom = hsum16(denom);
      const float inv = 1.0f / denom;
#pragma unroll
      for (int j = 0; j < 16; ++j) sv[j] *= inv;
    }
#pragma unroll
    for (int j = 0; j < 16; ++j) {
      const int c = sub + j * 16;
      sS[row][c] = sv[j];            // P for phase C
      attn_g[rbase + c] = sv[j];     // out_attn (coalesced)
    }
  }
  __syncthreads();

  // ================= Phase C1: out += P @ V via WMMA =================
  // P(16x256) @ V(256x64): K split across 8 waves (32 k each), 4 N-tiles of 16.
  // f32 WMMA 16x16x4 layouts (ISA 7.12.2):
  //   A 16x4 : lane L -> M = L&15, VGPR j -> K = j + 2*(L>>4)
  //   B 4x16 : lane L -> N = L&15, VGPR j -> K = j + 2*(L>>4)
  //   C 16x16: lane L -> N = L&15, VGPR r -> M = r + 8*(L>>4)
  {
    const int M = lane & 15;
    const int koff = (lane >> 4) * 2;
    v8f c0 = {}, c1 = {}, c2 = {}, c3 = {};
#pragma unroll
    for (int kc = 0; kc < 8; ++kc) {
      const int kb = wave * 32 + kc * 4;
      v2f a;
      a.x = sS[M][kb + koff];
      a.y = sS[M][kb + koff + 1];
      const float* vr0 = v_g + ((size_t)b * kLk + kb + koff) * kDk;  // row K
      const float* vr1 = vr0 + kDk;                                  // row K+1
      v2f b0, b1, b2, b3;
      b0.x = vr0[ 0 + M]; b0.y = vr1[ 0 + M];
      b1.x = vr0[16 + M]; b1.y = vr1[16 + M];
      b2.x = vr0[32 + M]; b2.y = vr1[32 + M];
      b3.x = vr0[48 + M]; b3.y = vr1[48 + M];
      c0 = __builtin_amdgcn_wmma_f32_16x16x4_f32(false, a, false, b0, (short)0, c0, false, false);
      c1 = __builtin_amdgcn_wmma_f32_16x16x4_f32(false, a, false, b1, (short)0, c1, false, false);
      c2 = __builtin_amdgcn_wmma_f32_16x16x4_f32(false, a, false, b2, (short)0, c2, false, false);
      c3 = __builtin_amdgcn_wmma_f32_16x16x4_f32(false, a, false, b3, (short)0, c3, false, false);
    }
    const int rbase = (lane >> 4) * 8;
#pragma unroll
    for (int r = 0; r < 8; ++r) {
      atomicAdd(&sO[rbase + r][ 0 + M], c0[r]);  // ds_add_f32
      atomicAdd(&sO[rbase + r][16 + M], c1[r]);
      atomicAdd(&sO[rbase + r][32 + M], c2[r]);
      atomicAdd(&sO[rbase + r][48 + M], c3[r]);
    }
  }

  // ================= Phase C2: out += P . v_pos (HBM stream #2) =================
  // Each half-wave owns one row i; lane owns 4 d-columns; accumulate in regs.
  {
    const int i  = wave * 2 + half;
    const int d0 = sub * 4;
    const float* prow =
        pos_g + (((size_t)b * kLq + q0 + i) * kLk) * (2 * kDk) + kDk;  // v_pos half
    float4 acc = {0.f, 0.f, 0.f, 0.f};
    for (int kk = 0; kk < 256; ++kk) {
      const float p = sS[i][kk];  // LDS broadcast within half-wave
      const float4 pv = *(const float4*)(prow + (size_t)kk * (2 * kDk) + d0);
      __builtin_prefetch(prow + (size_t)(kk + 4) * (2 * kDk) + d0, 0, 0);
      acc.x += p * pv.x; acc.y += p * pv.y;
      acc.z += p * pv.z; acc.w += p * pv.w;
    }
    atomicAdd(&sO[i][d0 + 0], acc.x);
    atomicAdd(&sO[i][d0 + 1], acc.y);
    atomicAdd(&sO[i][d0 + 2], acc.z);
    atomicAdd(&sO[i][d0 + 3], acc.w);
  }
  __syncthreads();

  // ---- write output tile (coalesced float4) ----
  {
    const int row = tid >> 4;
    const int col = (tid & 15) * 4;
    const float4 o = *(const float4*)&sO[row][col];
    *(float4*)&out_g[((size_t)b * kLq + q0 + row) * kDk + col] = o;
  }
}

extern "C" void kernel_launch(void* const* d_in, const int* in_sizes, int n_in,
                              void* d_out, int out_size, void* d_ws, size_t ws_size,
                              hipStream_t stream) {
  // setup_inputs order: q, k, v, attn_mask (bool -> 1 byte/elem), attn_pos_emb
  const float*   q    = (const float*)d_in[0];
  const float*   k    = (const float*)d_in[1];
  const float*   v    = (const float*)d_in[2];
  const uint8_t* mask = (const uint8_t*)d_in[3];
  const float*   pos  = (const float*)d_in[4];
  float* out  = (float*)d_out;                        // [16][256][64]
  float* attn = out + (size_t)kB * kLq * kDk;         // [16][256][256]
  dim3 grid(kLq / kQT, kB);                           // 16 x 16 = 256 WGs
  attn_pos_kernel<<<grid, 256, 0, stream>>>(q, k, v, mask, pos, out, attn);
  (void)in_sizes; (void)n_in; (void)out_size; (void)d_ws; (void)ws_size;
}